// RefLocal7_56710748176832
// MI455X (gfx1250) — compile-verified
//
#include <hip/hip_runtime.h>
#include <hip/hip_bf16.h>

typedef __attribute__((ext_vector_type(2))) float v2f;
typedef __attribute__((ext_vector_type(8))) float v8f;

#define BATCH 4
#define IMH   128
#define IMW   128
#define CH    64
#define NPIX  (BATCH * IMH * IMW)   // 65536 pixels
#define NSCORE 26                   // 25 window offsets + self

// ---------------------------------------------------------------------------
// Kernel 1: 1x1 conv == GEMM  Y[p,d] = sum_c X[p,c] * W[c,d]
// Uses V_WMMA_F32_16X16X4_F32 (full fp32 precision, CDNA5 matrix pipe).
// One wave -> 16 rows x 64 cols: 4 accumulators (v8f), 16 K-steps, 4 N-tiles.
// blockDim = 128 (4 waves) -> 64 rows per block.  gridDim.y selects main/ref.
// ---------------------------------------------------------------------------
__global__ __launch_bounds__(128) void conv1x1_wmma(
    const float* __restrict__ Xm, const float* __restrict__ Xr,
    const float* __restrict__ Wm, const float* __restrict__ Wr,
    float* __restrict__ ws)
{
    const float* X  = blockIdx.y ? Xr : Xm;
    const float* Wt = blockIdx.y ? Wr : Wm;
    float* Y = ws + (size_t)blockIdx.y * (size_t)NPIX * CH;

    const int wave = threadIdx.x >> 5;
    const int lane = threadIdx.x & 31;
    const int m = lane & 15;     // M row / N col within tile
    const int h = lane >> 4;     // lane half selector
    const int row0 = (blockIdx.x * 4 + wave) * 16;

    v8f acc0 = {}, acc1 = {}, acc2 = {}, acc3 = {};

    const float* xrow = X + (size_t)(row0 + m) * CH;

#pragma unroll
    for (int k = 0; k < 16; ++k) {
        const int kb = 4 * k + 2 * h;   // K_local = v + 2*h  (v = element idx)

        // A 16x4 fp32: VGPR v, lanes0-15 K=v, lanes16-31 K=v+2
        v2f a;
        a.x = xrow[kb];
        a.y = xrow[kb + 1];

        // B 4x16 fp32: VGPR v, lanes0-15 K=v, lanes16-31 K=v+2, N=lane&15
        const float* w0 = Wt + (size_t)kb * CH;
        v2f b0, b1, b2, b3;
        b0.x = w0[m];          b0.y = w0[CH + m];
        b1.x = w0[16 + m];     b1.y = w0[CH + 16 + m];
        b2.x = w0[32 + m];     b2.y = w0[CH + 32 + m];
        b3.x = w0[48 + m];     b3.y = w0[CH + 48 + m];

        acc0 = __builtin_amdgcn_wmma_f32_16x16x4_f32(false, a, false, b0, (short)0, acc0, false, false);
        acc1 = __builtin_amdgcn_wmma_f32_16x16x4_f32(false, a, false, b1, (short)0, acc1, false, false);
        acc2 = __builtin_amdgcn_wmma_f32_16x16x4_f32(false, a, false, b2, (short)0, acc2, false, false);
        acc3 = __builtin_amdgcn_wmma_f32_16x16x4_f32(false, a, false, b3, (short)0, acc3, false, false);
    }

    // C/D 16x16 f32: VGPR r, lanes0-15 M=r, lanes16-31 M=r+8
#pragma unroll
    for (int r = 0; r < 8; ++r) {
        const int row = row0 + r + 8 * h;
        float* y = Y + (size_t)row * CH + m;
        y[0]  = acc0[r];
        y[16] = acc1[r];
        y[32] = acc2[r];
        y[48] = acc3[r];
    }
}

// ---------------------------------------------------------------------------
// Kernel 2: per-pixel local-attention scores + softmax.
// scores[o=i*5+j] = dot(cm[p], cr[h+i-2, w+j-2])  (0 outside image, matching
// zero padding in the reference); scores[25] = dot(cm[p], cm[p]); softmax(26).
// ---------------------------------------------------------------------------
__global__ __launch_bounds__(256) void local_attn_softmax(
    const float* __restrict__ ws, float* __restrict__ out)
{
    const int p = blockIdx.x * blockDim.x + threadIdx.x;
    if (p >= NPIX) return;

    const int w = p & (IMW - 1);
    const int h = (p >> 7) & (IMH - 1);
    const int b = p >> 14;

    const float4* cm4 = (const float4*)(ws + (size_t)p * CH);
    const float* crbase = ws + (size_t)NPIX * CH;

    float4 q[CH / 4];
#pragma unroll
    for (int c = 0; c < CH / 4; ++c) q[c] = cm4[c];

    float scores[NSCORE];

    // self score (appended last by the reference)
    {
        float s = 0.f;
#pragma unroll
        for (int c = 0; c < CH / 4; ++c) {
            s = fmaf(q[c].x, q[c].x, s);
            s = fmaf(q[c].y, q[c].y, s);
            s = fmaf(q[c].z, q[c].z, s);
            s = fmaf(q[c].w, q[c].w, s);
        }
        scores[25] = s;
    }

#pragma unroll
    for (int i = 0; i < 5; ++i) {
#pragma unroll
        for (int j = 0; j < 5; ++j) {
            const int hh = h + i - 2;
            const int ww = w + j - 2;
            float s = 0.f;
            if (hh >= 0 && hh < IMH && ww >= 0 && ww < IMW) {
                const float4* k4 = (const float4*)(
                    crbase + (((size_t)b * IMH + hh) * IMW + ww) * CH);
#pragma unroll
                for (int c = 0; c < CH / 4; ++c) {
                    const float4 kv = k4[c];
                    s = fmaf(q[c].x, kv.x, s);
                    s = fmaf(q[c].y, kv.y, s);
                    s = fmaf(q[c].z, kv.z, s);
                    s = fmaf(q[c].w, kv.w, s);
                }
            }
            scores[i * 5 + j] = s;
        }
    }

    // softmax over 26 entries
    float mx = scores[0];
#pragma unroll
    for (int o = 1; o < NSCORE; ++o) mx = fmaxf(mx, scores[o]);
    float sum = 0.f;
#pragma unroll
    for (int o = 0; o < NSCORE; ++o) {
        const float e = __expf(scores[o] - mx);
        scores[o] = e;
        sum += e;
    }
    const float inv = 1.f / sum;
    float* op = out + (size_t)p * NSCORE;
#pragma unroll
    for (int o = 0; o < NSCORE; ++o) op[o] = scores[o] * inv;
}

// ---------------------------------------------------------------------------
extern "C" void kernel_launch(void* const* d_in, const int* in_sizes, int n_in,
                              void* d_out, int out_size, void* d_ws, size_t ws_size,
                              hipStream_t stream) {
    (void)in_sizes; (void)n_in; (void)out_size; (void)ws_size;
    const float* xm = (const float*)d_in[0];
    const float* xr = (const float*)d_in[1];
    const float* wm = (const float*)d_in[2];
    const float* wr = (const float*)d_in[3];
    float* out = (float*)d_out;
    float* ws  = (float*)d_ws;   // cm at [0, NPIX*CH), cr at [NPIX*CH, 2*NPIX*CH)

    dim3 gridA(NPIX / 64, 2, 1);          // 64 rows per block, y: main/ref
    conv1x1_wmma<<<gridA, 128, 0, stream>>>(xm, xr, wm, wr, ws);

    local_attn_softmax<<<(NPIX + 255) / 256, 256, 0, stream>>>(ws, out);
}